// DSMv3_44667659878435
// MI455X (gfx1250) — compile-verified
//
#include <hip/hip_runtime.h>

// Modulated deformable sampling (DCNv2 gather stage) for MI455X / gfx1250.
// B=8, C=256, H=W=64, G=4 deformable groups, K=3x3, stride=1, pad=1, dil=1.
//
// Roofline: ~81 MB compulsory traffic -> 3.5 us HBM floor; 1.2 GB of 4B
// gathers served from L0/L2 (input 33.5 MB << 192 MB L2). Gather-bandwidth
// bound; no matmul structure (gather index varies per output column), so the
// optimal path is maximal in-flight coalesced/gather vmem, not WMMA.

#define HW   4096   // H*W
#define CG   64     // channels per group
#define KK   9
#define CPT  32     // channels per thread (2 threads split one (pixel,group))

__global__ __launch_bounds__(256) void dcn_sample_kernel(
    const float* __restrict__ input,   // [8,256,64,64]
    const float* __restrict__ offset,  // [8,72,64,64]  ch = (g*9+k)*2 + {dy,dx}
    const float* __restrict__ mask,    // [8,36,64,64]  ch = g*9+k
    float* __restrict__ out)           // [8,256,64,64]
{
    const unsigned t = blockIdx.x * 256u + threadIdx.x;   // [0, 262144)
    const int wo = t & 63;              // lanes 0..31 -> consecutive wo: coalesced
    const int ho = (t >> 6) & 63;
    const int chalf = (t >> 12) & 1;    // which half of the 64 group-channels
    const int g  = (t >> 13) & 3;
    const int b  = (int)(t >> 15);
    const int p  = ho * 64 + wo;

    const float* __restrict__ xg = input + ((((b << 2) | g) * CG) + chalf * CPT) * HW;
    const float* __restrict__ offp = offset + (b * 72 + g * 18) * HW + p;
    const float* __restrict__ mp   = mask   + (b * 36 + g * 9)  * HW + p;
    float* __restrict__ op = out + ((((b << 2) | g) * CG) + chalf * CPT) * HW + p;

    float acc[CPT];
#pragma unroll
    for (int c = 0; c < CPT; ++c) acc[c] = 0.f;

    // Software-pipelined tap parameters: load tap k+1's (dy,dx,m) BEFORE
    // issuing tap k's 128 gathers, so the parameter loads (in-order LOADcnt
    // stream) overlap the gather phase instead of serializing each tap.
    float dy = offp[0];
    float dx = offp[HW];
    float m  = mp[0];

    for (int k = 0; k < KK; ++k) {
        float dy_n = 0.f, dx_n = 0.f, m_n = 0.f;
        if (k + 1 < KK) {
            dy_n = offp[(2 * k + 2) * HW];
            dx_n = offp[(2 * k + 3) * HW];
            m_n  = mp[(k + 1) * HW];
        }

        // Sampling position: base grid + tap + learned offset.
        const float py = (float)(ho - 1 + (k / 3)) + dy;
        const float px = (float)(wo - 1 + (k % 3)) + dx;
        const float y0f = floorf(py);
        const float x0f = floorf(px);
        const int y0 = (int)y0f;
        const int x0 = (int)x0f;
        const float ly = py - y0f;
        const float lx = px - x0f;
        const float hy = 1.f - ly;
        const float hx = 1.f - lx;
        const int y1 = y0 + 1, x1 = x0 + 1;

        const bool vy0 = ((unsigned)y0 < 64u);
        const bool vy1 = ((unsigned)y1 < 64u);
        const bool vx0 = ((unsigned)x0 < 64u);
        const bool vx1 = ((unsigned)x1 < 64u);

        // Bilinear weights x mask x validity, computed ONCE per tap,
        // amortized over CPT channels.
        const float w00 = (vy0 && vx0) ? hy * hx * m : 0.f;
        const float w01 = (vy0 && vx1) ? hy * lx * m : 0.f;
        const float w10 = (vy1 && vx0) ? ly * hx * m : 0.f;
        const float w11 = (vy1 && vx1) ? ly * lx * m : 0.f;

        const int cy0 = min(max(y0, 0), 63);
        const int cy1 = min(max(y1, 0), 63);
        const int cx0 = min(max(x0, 0), 63);
        const int cx1 = min(max(x1, 0), 63);

        // Four per-lane gather bases; channel stride is a compile-time
        // immediate (c*16384 B fits the signed 24-bit IOFFSET), so the
        // unrolled loop issues 128 independent global_load_b32 per tap.
        const float* __restrict__ p00 = xg + cy0 * 64 + cx0;
        const float* __restrict__ p01 = xg + cy0 * 64 + cx1;
        const float* __restrict__ p10 = xg + cy1 * 64 + cx0;
        const float* __restrict__ p11 = xg + cy1 * 64 + cx1;

#pragma unroll
        for (int c = 0; c < CPT; ++c) {
            const int o = c * HW;
            float v = fmaf(p00[o], w00, acc[c]);
            v = fmaf(p01[o], w01, v);
            v = fmaf(p10[o], w10, v);
            v = fmaf(p11[o], w11, v);
            acc[c] = v;
        }

        dy = dy_n; dx = dx_n; m = m_n;
    }

    // Output is write-once, never re-read by this kernel: non-temporal stores.
#pragma unroll
    for (int c = 0; c < CPT; ++c) {
        __builtin_nontemporal_store(acc[c], op + c * HW);
    }
}

extern "C" void kernel_launch(void* const* d_in, const int* in_sizes, int n_in,
                              void* d_out, int out_size, void* d_ws, size_t ws_size,
                              hipStream_t stream) {
    const float* input  = (const float*)d_in[0];
    const float* offset = (const float*)d_in[1];
    const float* mask   = (const float*)d_in[2];
    float* out = (float*)d_out;

    // B*G*2*Ho*Wo = 8*4*2*64*64 = 262144 threads
    dim3 grid(1024), block(256);
    hipLaunchKernelGGL(dcn_sample_kernel, grid, block, 0, stream,
                       input, offset, mask, out);
}